// KGNN_37177236914932
// MI455X (gfx1250) — compile-verified
//
#include <hip/hip_runtime.h>
#include <hip/hip_bf16.h>

// ---------------------------------------------------------------------------
// GNN 3-layer conv for MI455X (gfx1250, wave32, WMMA).
// Per layer:
//   Y1 = H @ W1 ; Y2 = H @ W2          (v_wmma_f32_16x16x4_f32, exact fp32)
//   Y1[n] += sum_{j in CSR(n)} Y2[colS[j]]   (deterministic CSR gather; Y2
//                                             table is L2-resident: 51.2MB << 192MB)
//   BN (batch stats, biased var) + ReLU -> next H
// Key identity: x[col] @ W2 == (x @ W2)[col]  -> 278 GFLOP naive becomes
// ~33 GFLOP GEMM + ~2.6 GB of mostly-L2 gather traffic (23.3 TB/s HBM floor).
// ---------------------------------------------------------------------------

#define GN_N   50000
#define GN_E   800000
#define GN_CO  256

typedef __attribute__((ext_vector_type(2))) float v2f;
typedef __attribute__((ext_vector_type(8))) float v8f;

// ---------------------------------------------------------------------------
// Weight transpose: W [ci][co] -> Wt [co][ci], so each lane's B operand pair
// (B[kk][col], B[kk+1][col]) is one contiguous global_load_b64.
// ---------------------------------------------------------------------------
__global__ void transpose_w(const float* __restrict__ W, float* __restrict__ Wt,
                            int ci, int co)
{
    int i = blockIdx.x * blockDim.x + threadIdx.x;
    if (i < ci * co) {
        int r = i / co, c = i % co;
        Wt[(size_t)c * ci + r] = W[i];
    }
}

// ---------------------------------------------------------------------------
// Quad GEMM tile: one wave -> 16x16 tiles (m, n) and (m, n+8) of BOTH
// Y1 = H@W1 and Y2 = H@W2. Inner K=4 step: 1 A-load + 4 B-loads (all b64)
// + 4 v_wmma_f32_16x16x4_f32.
// Tiling: 3125 row-tiles x 8 n-pairs = 25000 wave-tiles, 8 waves/block.
// EXEC all-ones everywhere (exact tiling, no divergence) per WMMA rules.
// ---------------------------------------------------------------------------
__global__ void __launch_bounds__(256)
gemm_quad_wmma(const float* __restrict__ H,
               const float* __restrict__ W1t,   // [co][ci]
               const float* __restrict__ W2t,   // [co][ci]
               float* __restrict__ Y1,
               float* __restrict__ Y2,
               int ci, int co)
{
    const int wave   = threadIdx.x >> 5;
    const int lane   = threadIdx.x & 31;
    const int tile   = blockIdx.x * (blockDim.x >> 5) + wave;
    const int nPairs = co >> 5;              // 8
    const int m      = tile / nPairs;        // row tile
    const int np     = tile % nPairs;        // n-pair id
    const int laneLo = lane & 15;
    const int laneHi = lane >> 4;            // 0: K={0,1}, 1: K={2,3}

    const int bcol0 = np * 16 + laneLo;      // first col tile
    const int bcol1 = bcol0 + (co >> 1);     // second col tile (n+8)

    // A: row (m*16 + laneLo), two consecutive K values per lane per step.
    const float* __restrict__ arow = H + (size_t)(m * 16 + laneLo) * ci + 2 * laneHi;
    // B pointers: transposed weights, contiguous along K.
    const float* __restrict__ b1a = W1t + (size_t)bcol0 * ci + 2 * laneHi;
    const float* __restrict__ b1b = W1t + (size_t)bcol1 * ci + 2 * laneHi;
    const float* __restrict__ b2a = W2t + (size_t)bcol0 * ci + 2 * laneHi;
    const float* __restrict__ b2b = W2t + (size_t)bcol1 * ci + 2 * laneHi;

    v8f acc1a = {}, acc1b = {}, acc2a = {}, acc2b = {};

    #pragma unroll 4
    for (int k = 0; k < ci; k += 4) {
        v2f a   = *(const v2f*)(arow + k);   // 8B-aligned (ci, k even)
        v2f v1a = *(const v2f*)(b1a + k);
        v2f v1b = *(const v2f*)(b1b + k);
        v2f v2a = *(const v2f*)(b2a + k);
        v2f v2b = *(const v2f*)(b2b + k);
        acc1a = __builtin_amdgcn_wmma_f32_16x16x4_f32(false, a, false, v1a, (short)0, acc1a, false, false);
        acc1b = __builtin_amdgcn_wmma_f32_16x16x4_f32(false, a, false, v1b, (short)0, acc1b, false, false);
        acc2a = __builtin_amdgcn_wmma_f32_16x16x4_f32(false, a, false, v2a, (short)0, acc2a, false, false);
        acc2b = __builtin_amdgcn_wmma_f32_16x16x4_f32(false, a, false, v2b, (short)0, acc2b, false, false);
    }

    // C/D layout: VGPR v -> row m*16 + laneHi*8 + v.
    const size_t rbase = (size_t)(m * 16 + laneHi * 8);
    #pragma unroll
    for (int v = 0; v < 8; ++v) {
        const size_t r = (rbase + v) * co;
        Y1[r + bcol0] = acc1a[v];
        Y1[r + bcol1] = acc1b[v];
        Y2[r + bcol0] = acc2a[v];
        Y2[r + bcol1] = acc2b[v];
    }
}

// ---------------------------------------------------------------------------
// CSR build (once; reused by all 3 layers).
// ---------------------------------------------------------------------------
__global__ void zero_i32(int* __restrict__ p, int n)
{
    int i = blockIdx.x * blockDim.x + threadIdx.x;
    if (i < n) p[i] = 0;
}

__global__ void edge_count(const int* __restrict__ row, int* __restrict__ cnt, int E)
{
    int e = blockIdx.x * blockDim.x + threadIdx.x;
    if (e < E) atomicAdd(&cnt[row[e]], 1);
}

// Single-workgroup hierarchical exclusive scan of cnt[N] -> starts[N+1], cursor[N].
__global__ void __launch_bounds__(1024)
scan_counts(const int* __restrict__ cnt, int* __restrict__ starts,
            int* __restrict__ cursor, int N)
{
    __shared__ int lds[1024];
    const int t     = threadIdx.x;
    const int chunk = (N + 1023) / 1024;
    const int base  = t * chunk;

    int s = 0;
    for (int i = 0; i < chunk; ++i) {
        int g = base + i;
        if (g < N) s += cnt[g];
    }
    lds[t] = s;
    __syncthreads();
    if (t == 0) {
        int run = 0;
        for (int i = 0; i < 1024; ++i) { int tmp = lds[i]; lds[i] = run; run += tmp; }
        starts[N] = run;           // == E
    }
    __syncthreads();
    int run = lds[t];
    for (int i = 0; i < chunk; ++i) {
        int g = base + i;
        if (g < N) {
            int c = cnt[g];
            starts[g] = run;
            cursor[g] = run;
            run += c;
        }
    }
}

__global__ void edge_fill(const int* __restrict__ row, const int* __restrict__ col,
                          int* __restrict__ cursor, int* __restrict__ colS, int E)
{
    int e = blockIdx.x * blockDim.x + threadIdx.x;
    if (e < E) {
        int p = atomicAdd(&cursor[row[e]], 1);
        colS[p] = col[e];
    }
}

// ---------------------------------------------------------------------------
// Aggregation: Y1[n][c] += sum over CSR segment of Y2[src][c].
// One block (256 thr, c = threadIdx.x) per destination node. Deterministic
// float summation order per node; no float atomics.
// ---------------------------------------------------------------------------
__global__ void __launch_bounds__(GN_CO)
aggregate_csr(float* __restrict__ Y1, const float* __restrict__ Y2,
              const int* __restrict__ starts, const int* __restrict__ colS)
{
    const int n = blockIdx.x;
    const int c = threadIdx.x;
    float acc = Y1[(size_t)n * GN_CO + c];
    const int s = starts[n];
    const int e = starts[n + 1];
    for (int j = s; j < e; ++j) {
        const int src = colS[j];
        acc += Y2[(size_t)src * GN_CO + c];
    }
    Y1[(size_t)n * GN_CO + c] = acc;
}

// ---------------------------------------------------------------------------
// BatchNorm: two-stage deterministic column reduction (sum, sumsq).
// ---------------------------------------------------------------------------
__global__ void __launch_bounds__(GN_CO)
bn_partial(const float* __restrict__ Y, float* __restrict__ partial,
           int rowsPerChunk, int N)
{
    const int c  = threadIdx.x;
    const int b  = blockIdx.x;
    int r0 = b * rowsPerChunk;
    int r1 = r0 + rowsPerChunk;
    if (r1 > N) r1 = N;
    float s = 0.f, s2 = 0.f;
    for (int r = r0; r < r1; ++r) {
        float v = Y[(size_t)r * GN_CO + c];
        s  += v;
        s2 += v * v;
    }
    partial[(size_t)b * 2 * GN_CO + c]          = s;
    partial[(size_t)b * 2 * GN_CO + GN_CO + c]  = s2;
}

__global__ void __launch_bounds__(GN_CO)
bn_final(const float* __restrict__ partial, const float* __restrict__ gamma,
         const float* __restrict__ beta, float* __restrict__ stats,
         int nChunks, int N)
{
    const int c = threadIdx.x;
    float s = 0.f, s2 = 0.f;
    for (int b = 0; b < nChunks; ++b) {
        s  += partial[(size_t)b * 2 * GN_CO + c];
        s2 += partial[(size_t)b * 2 * GN_CO + GN_CO + c];
    }
    const float invN  = 1.0f / (float)N;
    const float mean  = s * invN;
    const float var   = s2 * invN - mean * mean;     // biased var (matches jnp)
    const float scale = gamma[c] * rsqrtf(var + 1e-5f);
    stats[c]          = scale;
    stats[GN_CO + c]  = beta[c] - mean * scale;
}

__global__ void norm_relu(const float* __restrict__ Y, const float* __restrict__ stats,
                          float* __restrict__ dst, size_t total)
{
    size_t i = (size_t)blockIdx.x * blockDim.x + threadIdx.x;
    if (i >= total) return;
    const int c = (int)(i & (GN_CO - 1));
    float v = Y[i] * stats[c] + stats[GN_CO + c];
    dst[i] = v > 0.f ? v : 0.f;
}

// ---------------------------------------------------------------------------
// Host-side orchestration
// ---------------------------------------------------------------------------
static inline size_t alignup(size_t x) { return (x + 255) & ~(size_t)255; }

extern "C" void kernel_launch(void* const* d_in, const int* in_sizes, int n_in,
                              void* d_out, int out_size, void* d_ws, size_t ws_size,
                              hipStream_t stream)
{
    (void)in_sizes; (void)n_in; (void)out_size; (void)ws_size;

    const float* x    = (const float*)d_in[0];
    const int*   ei   = (const int*)d_in[1];
    const int*   row  = ei;            // edge_index[0]
    const int*   col  = ei + GN_E;     // edge_index[1]
    const float* W1[3] = { (const float*)d_in[2], (const float*)d_in[6],  (const float*)d_in[10] };
    const float* W2[3] = { (const float*)d_in[3], (const float*)d_in[7],  (const float*)d_in[11] };
    const float* G [3] = { (const float*)d_in[4], (const float*)d_in[8],  (const float*)d_in[12] };
    const float* B [3] = { (const float*)d_in[5], (const float*)d_in[9],  (const float*)d_in[13] };

    // Workspace carve-up
    char* w = (char*)d_ws;
    size_t off = 0;
    const size_t featBytes = (size_t)GN_N * GN_CO * sizeof(float); // 51.2 MB
    float* P0 = (float*)(w + off); off += alignup(featBytes);      // Y1 / out
    float* P1 = (float*)(w + off); off += alignup(featBytes);      // Y2
    float* P2 = (float*)(w + off); off += alignup(featBytes);      // normalized H
    int* colS   = (int*)(w + off); off += alignup((size_t)GN_E * 4);
    int* cnt    = (int*)(w + off); off += alignup((size_t)GN_N * 4);
    int* starts = (int*)(w + off); off += alignup((size_t)(GN_N + 1) * 4);
    int* cursor = (int*)(w + off); off += alignup((size_t)GN_N * 4);
    const int nChunks = 250, rowsPerChunk = 200;                   // 250*200 = N
    float* partial = (float*)(w + off); off += alignup((size_t)nChunks * 2 * GN_CO * 4);
    float* stats   = (float*)(w + off); off += alignup((size_t)2 * GN_CO * 4);
    float* W1t     = (float*)(w + off); off += alignup((size_t)GN_CO * GN_CO * 4);
    float* W2t     = (float*)(w + off); off += alignup((size_t)GN_CO * GN_CO * 4);

    // ---- CSR build (once, reused by all layers) ----
    zero_i32<<<(GN_N + 255) / 256, 256, 0, stream>>>(cnt, GN_N);
    edge_count<<<(GN_E + 255) / 256, 256, 0, stream>>>(row, cnt, GN_E);
    scan_counts<<<1, 1024, 0, stream>>>(cnt, starts, cursor, GN_N);
    edge_fill<<<(GN_E + 255) / 256, 256, 0, stream>>>(row, col, cursor, colS, GN_E);

    const size_t total = (size_t)GN_N * GN_CO;

    const float* Hin = x;
    int ci = 128;
    for (int L = 0; L < 3; ++L) {
        float* dst = (L == 2) ? (float*)d_out : P2;

        // Pre-transpose this layer's weights (tiny: ci*co*4 bytes each).
        const int wElems  = ci * GN_CO;
        transpose_w<<<(wElems + 255) / 256, 256, 0, stream>>>(W1[L], W1t, ci, GN_CO);
        transpose_w<<<(wElems + 255) / 256, 256, 0, stream>>>(W2[L], W2t, ci, GN_CO);

        // 3125 row-tiles * 8 n-pairs = 25000 wave-tiles, 8 waves per block.
        const int waveTiles  = (GN_N / 16) * (GN_CO / 32);
        const int gemmBlocks = waveTiles / 8;
        gemm_quad_wmma<<<gemmBlocks, 256, 0, stream>>>(Hin, W1t, W2t, P0, P1, ci, GN_CO);

        aggregate_csr<<<GN_N, GN_CO, 0, stream>>>(P0, P1, starts, colS);
        bn_partial<<<nChunks, GN_CO, 0, stream>>>(P0, partial, rowsPerChunk, GN_N);
        bn_final<<<1, GN_CO, 0, stream>>>(partial, G[L], B[L], stats, nChunks, GN_N);
        norm_relu<<<(int)(total / 256), 256, 0, stream>>>(P0, stats, dst, total);

        Hin = P2;
        ci  = GN_CO;
    }
}